// GCN_20950850470456
// MI455X (gfx1250) — compile-verified
//
#include <hip/hip_runtime.h>
#include <hip/hip_bf16.h>

typedef __attribute__((ext_vector_type(2))) float v2f;
typedef __attribute__((ext_vector_type(8))) float v8f;

#define N_NODES   100000
#define N_EDGES   3200000
#define IN_FEATS  128
#define HIDDEN    16
#define NUM_CLS   10
#define NTILES    (N_NODES / 16)   // 6250 exactly

// ---------------- degree accumulation ----------------
__global__ void gcn_deg_kernel(const int* __restrict__ src, const int* __restrict__ dst,
                               unsigned int* __restrict__ outdeg, unsigned int* __restrict__ indeg,
                               int nedges) {
    int e = blockIdx.x * blockDim.x + threadIdx.x;
    if (e >= nedges) return;
    atomicAdd(&outdeg[src[e]], 1u);
    atomicAdd(&indeg[dst[e]], 1u);
}

// ---------------- per-node norms ----------------
__global__ void gcn_norm_kernel(const unsigned int* __restrict__ outdeg,
                                const unsigned int* __restrict__ indeg,
                                float* __restrict__ src_norm, float* __restrict__ dst_norm, int n) {
    int i = blockIdx.x * blockDim.x + threadIdx.x;
    if (i >= n) return;
    src_norm[i] = rsqrtf(fmaxf((float)outdeg[i], 1.0f));
    dst_norm[i] = rsqrtf(fmaxf((float)indeg[i], 1.0f));
}

// ---------------- layer-1 GEMM: h1 = (x * src_norm) @ W1, via V_WMMA_F32_16X16X4_F32 ----------------
// block = 128 threads = 4 waves; one 16-row tile per wave; K = 128 in 32 wmma steps.
__global__ void __launch_bounds__(128)
gcn_gemm1_wmma(const float* __restrict__ x, const float* __restrict__ W1,
               const float* __restrict__ src_norm, float* __restrict__ h1, int ntiles) {
    __shared__ float ldsW[IN_FEATS * 16];        // W1 transposed-friendly: [k*16 + n]
    __shared__ float ldsX[4][16 * IN_FEATS];     // x tile transposed:      [k*16 + row]
    const int lane = threadIdx.x & 31;
    const int w    = threadIdx.x >> 5;
    const int tile = blockIdx.x * 4 + w;

    // stage W1 (row-major [128][16] already matches [k*16+n])
    for (int i = threadIdx.x; i < IN_FEATS * 16; i += 128) ldsW[i] = W1[i];

    // stage x tile, transposed + pre-scaled by src_norm
    if (tile < ntiles) {
        const int rowbase = tile * 16;
        #pragma unroll 4
        for (int i = 0; i < 64; ++i) {
            int idx = lane + i * 32;          // 0..2047
            int r = idx >> 7;                 // row in tile
            int k = idx & 127;                // feature
            ldsX[w][k * 16 + r] = x[(rowbase + r) * IN_FEATS + k] * src_norm[rowbase + r];
        }
    }
    __syncthreads();
    if (tile >= ntiles) return;               // wave-uniform: EXEC stays all-ones for WMMA

    const int hi = lane >> 4, lo = lane & 15;
    v8f c = {};
    const float* Xw = &ldsX[w][0];
    #pragma unroll
    for (int k0 = 0; k0 < IN_FEATS; k0 += 4) {
        int base = (k0 + 2 * hi) * 16 + lo;
        v2f a, b;
        a.x = Xw[base];       a.y = Xw[base + 16];
        b.x = ldsW[base];     b.y = ldsW[base + 16];
        c = __builtin_amdgcn_wmma_f32_16x16x4_f32(false, a, false, b, (short)0, c, false, false);
    }
    const int rowbase = tile * 16;
    #pragma unroll
    for (int v = 0; v < 8; ++v)
        h1[(rowbase + v + 8 * hi) * 16 + lo] = c[v];
}

// ---------------- layer-2 GEMM: h2 = (relu_out * src_norm) @ W2pad (N padded 10->16) ----------------
__global__ void __launch_bounds__(128)
gcn_gemm2_wmma(const float* __restrict__ h, const float* __restrict__ W2,
               const float* __restrict__ src_norm, float* __restrict__ h2, int ntiles) {
    __shared__ float ldsW[16 * 16];
    __shared__ float ldsX[4][16 * 16];
    const int lane = threadIdx.x & 31;
    const int w    = threadIdx.x >> 5;
    const int tile = blockIdx.x * 4 + w;

    if (threadIdx.x < 16 * 16) {
        int k = threadIdx.x >> 4, n = threadIdx.x & 15;
        ldsW[threadIdx.x] = (n < NUM_CLS) ? W2[k * NUM_CLS + n] : 0.0f;
    }
    if (tile < ntiles) {
        const int rowbase = tile * 16;
        #pragma unroll
        for (int i = 0; i < 8; ++i) {
            int idx = lane + i * 32;          // 0..255
            int r = idx >> 4;
            int k = idx & 15;
            ldsX[w][k * 16 + r] = h[(rowbase + r) * 16 + k] * src_norm[rowbase + r];
        }
    }
    __syncthreads();
    if (tile >= ntiles) return;

    const int hi = lane >> 4, lo = lane & 15;
    v8f c = {};
    #pragma unroll
    for (int k0 = 0; k0 < 16; k0 += 4) {
        int base = (k0 + 2 * hi) * 16 + lo;
        v2f a, b;
        a.x = ldsX[w][base];  a.y = ldsX[w][base + 16];
        b.x = ldsW[base];     b.y = ldsW[base + 16];
        c = __builtin_amdgcn_wmma_f32_16x16x4_f32(false, a, false, b, (short)0, c, false, false);
    }
    const int rowbase = tile * 16;
    #pragma unroll
    for (int v = 0; v < 8; ++v)
        h2[(rowbase + v + 8 * hi) * 16 + lo] = c[v];
}

// ---------------- edge scatter-add: agg[dst] += h[src], 16 feats, 16 lanes/edge ----------------
__global__ void gcn_edge_agg(const float* __restrict__ h, const int* __restrict__ src,
                             const int* __restrict__ dst, float* __restrict__ agg, int nwork) {
    int t = blockIdx.x * blockDim.x + threadIdx.x;
    if (t >= nwork) return;
    int e = t >> 4;
    int f = t & 15;
    atomicAdd(&agg[dst[e] * 16 + f], h[src[e] * 16 + f]);
}

// ---------------- layer-1 epilogue: relu(agg*dst_norm + b1), in place ----------------
__global__ void gcn_finalize1(float* __restrict__ agg, const float* __restrict__ dst_norm,
                              const float* __restrict__ b1, int n16) {
    int i = blockIdx.x * blockDim.x + threadIdx.x;
    if (i >= n16) return;
    int node = i >> 4, f = i & 15;
    float v = agg[i] * dst_norm[node] + b1[f];
    agg[i] = fmaxf(v, 0.0f);
}

// ---------------- final output: out[N][10] = agg2*dst_norm + b2 ----------------
__global__ void gcn_final_out(const float* __restrict__ agg2, const float* __restrict__ dst_norm,
                              const float* __restrict__ b2, float* __restrict__ out, int n10) {
    int i = blockIdx.x * blockDim.x + threadIdx.x;
    if (i >= n10) return;
    int node = i / NUM_CLS;
    int c = i - node * NUM_CLS;
    out[i] = agg2[node * 16 + c] * dst_norm[node] + b2[c];
}

extern "C" void kernel_launch(void* const* d_in, const int* in_sizes, int n_in,
                              void* d_out, int out_size, void* d_ws, size_t ws_size,
                              hipStream_t stream) {
    const float* x  = (const float*)d_in[0];
    const int*   ei = (const int*)d_in[1];          // edge_index [2][E]
    const float* W1 = (const float*)d_in[2];
    const float* b1 = (const float*)d_in[3];
    const float* W2 = (const float*)d_in[4];
    const float* b2 = (const float*)d_in[5];
    float* out = (float*)d_out;

    const int* src = ei;
    const int* dst = ei + N_EDGES;

    // workspace layout
    char* ws = (char*)d_ws;
    unsigned int* outdeg = (unsigned int*)(ws);                       // N u32
    unsigned int* indeg  = (unsigned int*)(ws + (size_t)N_NODES * 4); // N u32
    float* src_norm = (float*)(ws + (size_t)N_NODES * 8);
    float* dst_norm = (float*)(ws + (size_t)N_NODES * 12);
    float* h1   = (float*)(ws + (size_t)N_NODES * 16);                      // N*16
    float* agg1 = (float*)(ws + (size_t)N_NODES * 16 + (size_t)N_NODES * 64);
    float* h2   = (float*)(ws + (size_t)N_NODES * 16 + (size_t)N_NODES * 128);
    float* agg2 = (float*)(ws + (size_t)N_NODES * 16 + (size_t)N_NODES * 192);

    // zero accumulators (graph-capturable memset nodes; re-done every call for determinism)
    hipMemsetAsync(outdeg, 0, (size_t)N_NODES * 4, stream);
    hipMemsetAsync(indeg,  0, (size_t)N_NODES * 4, stream);
    hipMemsetAsync(agg1,   0, (size_t)N_NODES * 64, stream);
    hipMemsetAsync(agg2,   0, (size_t)N_NODES * 64, stream);

    // degrees + norms
    gcn_deg_kernel<<<(N_EDGES + 255) / 256, 256, 0, stream>>>(src, dst, outdeg, indeg, N_EDGES);
    gcn_norm_kernel<<<(N_NODES + 255) / 256, 256, 0, stream>>>(outdeg, indeg, src_norm, dst_norm, N_NODES);

    // layer 1
    gcn_gemm1_wmma<<<(NTILES + 3) / 4, 128, 0, stream>>>(x, W1, src_norm, h1, NTILES);
    {
        int nwork = N_EDGES * 16;
        gcn_edge_agg<<<(nwork + 255) / 256, 256, 0, stream>>>(h1, src, dst, agg1, nwork);
    }
    gcn_finalize1<<<(N_NODES * 16 + 255) / 256, 256, 0, stream>>>(agg1, dst_norm, b1, N_NODES * 16);

    // layer 2 (agg1 now holds relu output)
    gcn_gemm2_wmma<<<(NTILES + 3) / 4, 128, 0, stream>>>(agg1, W2, src_norm, h2, NTILES);
    {
        int nwork = N_EDGES * 16;
        gcn_edge_agg<<<(nwork + 255) / 256, 256, 0, stream>>>(h2, src, dst, agg2, nwork);
    }
    gcn_final_out<<<(N_NODES * NUM_CLS + 255) / 256, 256, 0, stream>>>(agg2, dst_norm, b2, out, N_NODES * NUM_CLS);
}